// MultiLayerLSTM_40312563040982
// MI455X (gfx1250) — compile-verified
//
#include <hip/hip_runtime.h>
#include <cstdint>
#include <cstddef>

// ---------------------------------------------------------------------------
// 2-layer LSTM LM forward for MI455X (gfx1250, wave32, WMMA).
//  - f32 WMMA (V_WMMA_F32_16X16X4_F32) everywhere: reference is f32 and the
//    512-step recurrence would compound bf16 error.
//  - Sequential part: persistent 16-block kernel, 64 waves; each wave owns 16
//    hidden units (all 4 gates, cell state in VGPRs), weights pre-packed into
//    WMMA B-fragment order for coalesced b64 streaming from L2.
//  - A rows 8..15 (batch padding) are never zeroed: WMMA rows are independent,
//    garbage only reaches D rows 8..15 which are never stored. Lanes m>=8
//    alias lane m&7's LDS address (broadcast read, no exec masking).
//  - Logits: batched [4096x1024]x[1024x32000] LDS-staged WMMA GEMM.
// ---------------------------------------------------------------------------

typedef float v2f __attribute__((ext_vector_type(2)));
typedef float v8f __attribute__((ext_vector_type(8)));

#define VOCAB  32000
#define DMODEL 1024
#define NL     2
#define BB     8
#define TT     512
#define G4     (4 * DMODEL)   // 4096 gate columns
#define NBLK   16             // persistent blocks for recurrent kernel
#define NWAVE  4              // waves per block (128 threads)

__device__ __forceinline__ float sigmf_(float x) {
    return 1.0f / (1.0f + __expf(-x));
}

__device__ __forceinline__ v8f wmma_f32(v2f a, v2f b, v8f c) {
    // D = A(16x4 f32) * B(4x16 f32) + C(16x16 f32)
    return __builtin_amdgcn_wmma_f32_16x16x4_f32(
        /*neg_a=*/false, a, /*neg_b=*/false, b,
        /*c_mod=*/(short)0, c, /*reuse_a=*/false, /*reuse_b=*/false);
}

// ---------------------------------------------------------------------------
// Kernel 0: zero barrier counter + hidden-state double buffer (deterministic
// per launch; harness does not re-poison between replays).
// ---------------------------------------------------------------------------
__global__ void init_ws(unsigned* __restrict__ cnt, float* __restrict__ hst) {
    int tid = blockIdx.x * blockDim.x + threadIdx.x;
    if (tid == 0) *cnt = 0u;
    const int n = 2 * NL * BB * DMODEL;
    for (int i = tid; i < n; i += gridDim.x * blockDim.x) hst[i] = 0.0f;
}

// ---------------------------------------------------------------------------
// Kernel 1: pack stacked gate weights [wx;wh] (K=2048 x 4096, row-major) into
// WMMA B-fragment lane order:
//   Wpk[ ((((l*64+u)*4+g)*512+s)*32+lane ] = { W[4s+2*(lane>>4)  ][col],
//                                              W[4s+2*(lane>>4)+1][col] }
//   col = g*1024 + u*16 + (lane&15)
// -> inner GEMM loads one contiguous float2 per lane per WMMA.
// ---------------------------------------------------------------------------
__global__ void pack_w(const float* __restrict__ wx,
                       const float* __restrict__ wh,
                       float2* __restrict__ Wpk) {
    size_t id = (size_t)blockIdx.x * blockDim.x + threadIdx.x;
    const size_t total = (size_t)NL * 64 * 4 * 512 * 32;  // 2^23
    if (id >= total) return;
    int ln = (int)(id & 31);
    int s  = (int)((id >> 5) & 511);
    int g  = (int)((id >> 14) & 3);
    int u  = (int)((id >> 16) & 63);
    int l  = (int)(id >> 22);
    int col = g * DMODEL + u * 16 + (ln & 15);
    int k0  = 4 * s + 2 * (ln >> 4);

    float e0, e1;
    if (k0 < DMODEL) {
        const float* base = wx + ((size_t)l * DMODEL + k0) * G4 + col;
        e0 = base[0];
        e1 = base[G4];
    } else {
        const float* base = wh + ((size_t)l * DMODEL + (k0 - DMODEL)) * G4 + col;
        e0 = base[0];
        e1 = base[G4];
    }
    float2 v; v.x = e0; v.y = e1;
    Wpk[id] = v;
}

// ---------------------------------------------------------------------------
// Kernel 2: persistent recurrent kernel. 16 blocks x 128 threads = 64 waves.
// Wave w of block b owns hidden units [u*16, u*16+16), u = b*4+w, both layers.
// Per (t, layer):  A = [x_input (K 0..1023) ; h_prev (K 1024..2047)] staged in
// LDS quad-packed [K/4][8] (b128 fill, b64 frag reads); gates via 512x4 WMMA
// f32 16x16x4; elementwise LSTM update with cell state in registers; grid
// barrier (atomic) between phases.
// ---------------------------------------------------------------------------
__global__ __launch_bounds__(128) void lstm_seq(
    const int*    __restrict__ x,
    const float*  __restrict__ emb,
    const float*  __restrict__ bx,
    const float*  __restrict__ bh,
    const float2* __restrict__ Wpk,
    float*        __restrict__ hst,   // [parity][layer][B][D]
    float*        __restrict__ H2,    // [T][B][D] layer-1 outputs
    unsigned*     __restrict__ cnt) {
    // Quad-packed A: Alds4[k>>2][m] = {A[m][4q..4q+3]}, m<8. 64 KB.
    __shared__ float4 Alds4[(2 * DMODEL / 4) * BB];

    const int tid  = threadIdx.x;
    const int lane = tid & 31;
    const int wid  = tid >> 5;
    const int u    = blockIdx.x * NWAVE + wid;  // 0..63
    const int m16  = lane & 15;
    const int m8   = lane & 7;                  // aliased row for padding lanes
    const int hsel = lane >> 4;                 // 0: K lo pair, 1: K hi pair
    const int col  = u * 16 + m16;              // hidden unit this lane covers

    // Per-column gate bias (bx+bh), constant over time.
    float biasv[NL][4];
#pragma unroll
    for (int l = 0; l < NL; ++l)
#pragma unroll
        for (int g = 0; g < 4; ++g) {
            int c = l * G4 + g * DMODEL + col;
            biasv[l][g] = bx[c] + bh[c];
        }

    // Cell state in registers: creg[l][j] <-> (batch row, unit col) of D tile.
    float creg[NL][8];
#pragma unroll
    for (int l = 0; l < NL; ++l)
#pragma unroll
        for (int j = 0; j < 8; ++j) creg[l][j] = 0.0f;

    const float2* WbL0 = Wpk + (size_t)(0 * 64 + u) * 65536 + lane;
    const float2* WbL1 = Wpk + (size_t)(1 * 64 + u) * 65536 + lane;
    const float2* Ap   = (const float2*)Alds4;

    const int fm = tid >> 4;  // fill: batch row 0..7
    const int fj = tid & 15;  // fill: 16 threads per row

    unsigned bar_target = 0;

    for (int t = 0; t < TT; ++t) {
        const int p = t & 1;
#pragma unroll
        for (int l = 0; l < NL; ++l) {
            // ------- stage A into LDS (quad-packed [K/4][8]) -------
            const float* srcIn;
            if (l == 0) {
                int xi = x[fm * TT + t];
                srcIn = emb + (size_t)xi * DMODEL;
            } else {
                srcIn = hst + ((size_t)(p * NL + 0) * BB + fm) * DMODEL;
            }
            const float* srcH =
                hst + ((size_t)((p ^ 1) * NL + l) * BB + fm) * DMODEL;
            for (int q = fj; q < 256; q += 16) {
                Alds4[q * BB + fm] = *(const float4*)(srcIn + 4 * q);
            }
            for (int q = fj; q < 256; q += 16) {
                Alds4[(256 + q) * BB + fm] = *(const float4*)(srcH + 4 * q);
            }
            __syncthreads();

            // ------- gates: 4 accumulators, K=2048 in steps of 4 -------
            v8f acc[4];
#pragma unroll
            for (int g = 0; g < 4; ++g)
#pragma unroll
                for (int j = 0; j < 8; ++j) acc[g][j] = biasv[l][g];

            const float2* Wb = (l == 0) ? WbL0 : WbL1;
#pragma unroll 8
            for (int s = 0; s < 512; ++s) {
                // A frag: quad s, pair hsel, row m8 (lanes m16>=8 broadcast
                // lane m8's data; resulting D rows 8..15 are never stored).
                float2 av = Ap[(s * BB + m8) * 2 + hsel];
                v2f a; a[0] = av.x; a[1] = av.y;
#pragma unroll
                for (int g = 0; g < 4; ++g) {
                    float2 bv = Wb[(g * 512 + s) * 32];
                    v2f b; b[0] = bv.x; b[1] = bv.y;
                    acc[g] = wmma_f32(a, b, acc[g]);
                }
            }

            // ------- elementwise LSTM update + publish h -------
            float* hdst = hst + (size_t)(p * NL + l) * BB * DMODEL;
#pragma unroll
            for (int j = 0; j < 8; ++j) {
                float f  = sigmf_(acc[0][j]);
                float i  = sigmf_(acc[1][j]);
                float gg = tanhf(acc[2][j]);
                float o  = sigmf_(acc[3][j]);
                float cn = f * creg[l][j] + i * gg;
                creg[l][j] = cn;
                float h = o * tanhf(cn);
                if (hsel == 0) {                       // valid rows m=j (0..7)
                    hdst[(size_t)j * DMODEL + col] = h;
                    if (l == 1)
                        H2[((size_t)t * BB + j) * DMODEL + col] = h;
                }
            }

            // ------- grid barrier (16 blocks) -------
            bar_target += NBLK;
            __syncthreads();
            __threadfence();
            if (tid == 0) {
                __hip_atomic_fetch_add(cnt, 1u, __ATOMIC_RELEASE,
                                       __HIP_MEMORY_SCOPE_AGENT);
                while (__hip_atomic_load(cnt, __ATOMIC_ACQUIRE,
                                         __HIP_MEMORY_SCOPE_AGENT) < bar_target) {
                    __builtin_amdgcn_s_sleep(2);
                }
            }
            __syncthreads();
        }
    }
}

// ---------------------------------------------------------------------------
// Kernel 3: logits = H2[4096x1024] @ emb^T[1024x32000] + by, stored [B][T][V].
// Block tile 128(M) x 64(N), 8 waves (one 16-row M tile each, 4 N tiles),
// K staged in 32-wide quad-packed LDS panels (A: 16KB, B: 8KB; b128 fills).
// ---------------------------------------------------------------------------
__global__ __launch_bounds__(256) void logits_gemm(
    const float* __restrict__ H2,
    const float* __restrict__ emb,
    const float* __restrict__ by,
    float*       __restrict__ out) {
    __shared__ float4 Al4[8 * 128];  // [kquad][mm]
    __shared__ float4 Bl4[8 * 64];   // [kquad][nn]

    const int tid  = threadIdx.x;
    const int lane = tid & 31;
    const int wid  = tid >> 5;
    const int m16  = lane & 15;
    const int hsel = lane >> 4;
    const int n0   = blockIdx.x * 64;
    const int m0   = blockIdx.y * 128;

    v8f acc[4];
#pragma unroll
    for (int nt = 0; nt < 4; ++nt)
#pragma unroll
        for (int j = 0; j < 8; ++j) acc[nt][j] = 0.0f;

    const int amm = tid >> 1, ahalf = tid & 1;  // A fill: 128 rows x 2 halves
    const int bnn = tid >> 2, bq = tid & 3;     // B fill: 64 rows x 4 quarters

    const float2* Ap = (const float2*)Al4;
    const float2* Bp = (const float2*)Bl4;

    for (int kk = 0; kk < DMODEL; kk += 32) {
        __syncthreads();
        {
            const float* src =
                H2 + (size_t)(m0 + amm) * DMODEL + kk + ahalf * 16;
#pragma unroll
            for (int q = 0; q < 4; ++q) {           // 4 quads per half
                Al4[(ahalf * 4 + q) * 128 + amm] = *(const float4*)(src + 4 * q);
            }
        }
        {
            const float* src =
                emb + (size_t)(n0 + bnn) * DMODEL + kk + bq * 8;
#pragma unroll
            for (int q = 0; q < 2; ++q) {           // 2 quads per quarter
                Bl4[(bq * 2 + q) * 64 + bnn] = *(const float4*)(src + 4 * q);
            }
        }
        __syncthreads();
#pragma unroll
        for (int s = 0; s < 8; ++s) {
            float2 av = Ap[(s * 128 + wid * 16 + m16) * 2 + hsel];
            v2f a; a[0] = av.x; a[1] = av.y;
#pragma unroll
            for (int nt = 0; nt < 4; ++nt) {
                float2 bvv = Bp[(s * 64 + nt * 16 + m16) * 2 + hsel];
                v2f b; b[0] = bvv.x; b[1] = bvv.y;
                acc[nt] = wmma_f32(a, b, acc[nt]);
            }
        }
    }

#pragma unroll
    for (int nt = 0; nt < 4; ++nt) {
        int v = n0 + nt * 16 + m16;
        float bias = by[v];
#pragma unroll
        for (int j = 0; j < 8; ++j) {
            int r  = m0 + wid * 16 + 8 * hsel + j;  // r = t*8 + b
            int tt = r >> 3;
            int b  = r & 7;
            out[((size_t)b * TT + tt) * VOCAB + v] = acc[nt][j] + bias;
        }
    }
}

// ---------------------------------------------------------------------------
extern "C" void kernel_launch(void* const* d_in, const int* in_sizes, int n_in,
                              void* d_out, int out_size, void* d_ws,
                              size_t ws_size, hipStream_t stream) {
    (void)in_sizes; (void)n_in; (void)out_size; (void)ws_size;
    const int*   x   = (const int*)d_in[0];
    const float* emb = (const float*)d_in[1];
    const float* wx  = (const float*)d_in[2];
    const float* wh  = (const float*)d_in[3];
    const float* bx  = (const float*)d_in[4];
    const float* bh  = (const float*)d_in[5];
    const float* by  = (const float*)d_in[6];
    float* out = (float*)d_out;

    // Workspace layout (floats): [cnt pad 64][hstate 32768][H2 4,194,304][Wpk]
    float*    wsf = (float*)d_ws;
    unsigned* cnt = (unsigned*)d_ws;
    float*    hst = wsf + 64;
    float*    H2  = hst + 2 * NL * BB * DMODEL;
    float2*   Wpk = (float2*)(H2 + (size_t)TT * BB * DMODEL);

    init_ws<<<64, 256, 0, stream>>>(cnt, hst);
    pack_w<<<32768, 256, 0, stream>>>(wx, wh, Wpk);
    lstm_seq<<<NBLK, 128, 0, stream>>>(x, emb, bx, bh, (const float2*)Wpk,
                                       hst, H2, cnt);
    logits_gemm<<<dim3(VOCAB / 64, (BB * TT) / 128), 256, 0, stream>>>(
        H2, emb, by, out);
}